// GraphDecoder_43018392437385
// MI455X (gfx1250) — compile-verified
//
#include <hip/hip_runtime.h>
#include <hip/hip_bf16.h>

#define DF   128          // feature dim
#define LDSP 132          // padded LDS row stride (floats) -> conflict-free A reads

typedef __attribute__((ext_vector_type(2))) float v2f;
typedef __attribute__((ext_vector_type(8))) float v8f;

// ---------------------------------------------------------------------------
// GCN normalization precompute
// ---------------------------------------------------------------------------
__global__ void k_deg_init(float* deg, int n) {
    int i = blockIdx.x * blockDim.x + threadIdx.x;
    if (i < n) deg[i] = 1.0f;                      // self-loop weight 1
}

__global__ void k_deg_edges(const int* __restrict__ dst,
                            const float* __restrict__ ew,
                            float* __restrict__ deg, int e) {
    int i = blockIdx.x * blockDim.x + threadIdx.x;
    if (i < e) unsafeAtomicAdd(&deg[dst[i]], ew[i]);
}

__global__ void k_dis(const float* __restrict__ deg,
                      float* __restrict__ dis,
                      float* __restrict__ selfn, int n) {
    int i = blockIdx.x * blockDim.x + threadIdx.x;
    if (i < n) {
        float d = deg[i];
        float r = (d > 0.0f) ? rsqrtf(fmaxf(d, 1e-12f)) : 0.0f;
        dis[i] = r;
        selfn[i] = r * r;                          // norm of the self-loop edge
    }
}

__global__ void k_norm(const int* __restrict__ src, const int* __restrict__ dst,
                       const float* __restrict__ ew, const float* __restrict__ dis,
                       float* __restrict__ norm, int e) {
    int i = blockIdx.x * blockDim.x + threadIdx.x;
    if (i < e) norm[i] = dis[src[i]] * ew[i] * dis[dst[i]];
}

// ---------------------------------------------------------------------------
// WMMA fp32 GEMM:  H[N x 128] = X[N x 128] * W[128 x 128]
// One block = 128 rows; 8 waves, wave w owns rows [16w,16w+16), loops 8 col tiles.
// ---------------------------------------------------------------------------
__global__ __launch_bounds__(256)
void k_gemm_wmma(const float* __restrict__ X, const float* __restrict__ W,
                 float* __restrict__ H, int nrows) {
    __shared__ float lx[128 * LDSP];
    __shared__ float lw[128 * LDSP];

    const int t       = threadIdx.x;
    const int rowbase = blockIdx.x * 128;

    // Cooperative float4 staging of the X row-tile and W into LDS.
    for (int it = 0; it < 16; ++it) {
        int idx = t + it * 256;            // 0..4095 float4 slots
        int r   = idx >> 5;                // row 0..127
        int c4  = (idx & 31) << 2;         // col 0,4,...,124
        int gr  = rowbase + r;
        float4 v = make_float4(0.f, 0.f, 0.f, 0.f);
        if (gr < nrows) v = *(const float4*)(X + (size_t)gr * DF + c4);
        *(float4*)(lx + r * LDSP + c4) = v;
        float4 wv = *(const float4*)(W + (size_t)r * DF + c4);
        *(float4*)(lw + r * LDSP + c4) = wv;
    }
    __syncthreads();

    const int wid  = t >> 5;               // wave 0..7
    const int lane = t & 31;
    const int lrow = lane & 15;            // M (or N) within tile
    const int hi   = lane >> 4;            // lane half selects K offset
    const float* ax = lx + (wid * 16 + lrow) * LDSP;

    for (int ct = 0; ct < 8; ++ct) {
        const int cb = ct * 16;
        v8f acc = {0.f, 0.f, 0.f, 0.f, 0.f, 0.f, 0.f, 0.f};
        #pragma unroll 8
        for (int k = 0; k < 128; k += 4) {
            const int ka = k + 2 * hi;
            v2f a, b;
            a.x = ax[ka];
            a.y = ax[ka + 1];
            b.x = lw[ka * LDSP + cb + lrow];
            b.y = lw[(ka + 1) * LDSP + cb + lrow];
            acc = __builtin_amdgcn_wmma_f32_16x16x4_f32(
                false, a, false, b, (short)0, acc, false, false);
        }
        const int col = cb + lrow;
        #pragma unroll
        for (int j = 0; j < 8; ++j) {
            int gr = rowbase + wid * 16 + j + hi * 8;   // C VGPR j -> M=j / M=j+8
            if (gr < nrows) H[(size_t)gr * DF + col] = acc[j];
        }
    }
}

// ---------------------------------------------------------------------------
// Aggregation: agg = selfloop + scatter-add of h[src]*norm into dst rows
// ---------------------------------------------------------------------------
__global__ void k_agg_init(const float* __restrict__ h,
                           const float* __restrict__ selfn,
                           float* __restrict__ agg, int nd) {
    int i = blockIdx.x * blockDim.x + threadIdx.x;
    if (i < nd) agg[i] = h[i] * selfn[i >> 7];
}

__global__ void k_agg_edges(const int* __restrict__ src, const int* __restrict__ dst,
                            const float* __restrict__ norm,
                            const float* __restrict__ h,
                            float* __restrict__ agg, int e) {
    int gid = blockIdx.x * blockDim.x + threadIdx.x;     // E*32 threads
    int ei  = gid >> 5;
    if (ei >= e) return;
    int q   = (gid & 31) << 2;                           // float4 slot in the row
    int s   = src[ei];
    int d   = dst[ei];
    float w = norm[ei];
    float4 v = *(const float4*)(h + (size_t)s * DF + q);
    float* p = agg + (size_t)d * DF + q;
    unsafeAtomicAdd(p + 0, v.x * w);
    unsafeAtomicAdd(p + 1, v.y * w);
    unsafeAtomicAdd(p + 2, v.z * w);
    unsafeAtomicAdd(p + 3, v.w * w);
}

// ---------------------------------------------------------------------------
// Post: bias (+ReLU) + skip, with fused per-column sum/sumsq for BatchNorm
// Grid-stride total threads must be a multiple of 128 so each thread owns one column.
// ---------------------------------------------------------------------------
__global__ void k_zero_stats(float* gsum, float* gsq) {
    int i = threadIdx.x;
    gsum[i] = 0.0f;
    gsq[i]  = 0.0f;
}

__global__ __launch_bounds__(256)
void k_post_z(const float* __restrict__ agg, const float* __restrict__ b,
              const float* __restrict__ x0, float* __restrict__ z,
              float* __restrict__ gsum, float* __restrict__ gsq,
              int nd, int do_relu) {
    int gt = blockIdx.x * blockDim.x + threadIdx.x;
    int d  = gt & 127;                 // fixed: stride is a multiple of 128
    float bd = b[d];
    float s = 0.0f, sq = 0.0f;
    for (int idx = gt; idx < nd; idx += gridDim.x * blockDim.x) {
        float v = agg[idx] + bd;
        if (do_relu) v = fmaxf(v, 0.0f);
        v += x0[idx];
        z[idx] = v;
        s  += v;
        sq += v * v;
    }
    unsafeAtomicAdd(&gsum[d], s);
    unsafeAtomicAdd(&gsq[d],  sq);
}

__global__ void k_bn_finalize(const float* __restrict__ gsum, const float* __restrict__ gsq,
                              const float* __restrict__ gamma, const float* __restrict__ beta,
                              float* __restrict__ scale, float* __restrict__ shift, int n) {
    int d = threadIdx.x;               // 128 threads
    float inv = 1.0f / (float)n;
    float mu  = gsum[d] * inv;
    float var = gsq[d] * inv - mu * mu;      // biased variance (ddof=0)
    float sc  = gamma[d] * rsqrtf(var + 1e-5f);
    scale[d] = sc;
    shift[d] = beta[d] - mu * sc;
}

__global__ void k_bn_apply(const float* __restrict__ z,
                           const float* __restrict__ scale, const float* __restrict__ shift,
                           float* __restrict__ out, int nd) {
    int i = blockIdx.x * blockDim.x + threadIdx.x;
    if (i < nd) {
        int d = i & 127;
        out[i] = z[i] * scale[d] + shift[d];
    }
}

__global__ void k_post_nobn(const float* __restrict__ agg, const float* __restrict__ b,
                            const float* __restrict__ x0, float* __restrict__ out, int nd) {
    int i = blockIdx.x * blockDim.x + threadIdx.x;
    if (i < nd) {
        float v = fmaxf(agg[i] + b[i & 127], 0.0f) + x0[i];   // layer 2: relu, no BN
        out[i] = v;
    }
}

// ---------------------------------------------------------------------------
extern "C" void kernel_launch(void* const* d_in, const int* in_sizes, int n_in,
                              void* d_out, int out_size, void* d_ws, size_t ws_size,
                              hipStream_t stream) {
    const float* x0 = (const float*)d_in[0];
    const int*   ei = (const int*)d_in[1];      // [2, E]: src then dst
    const float* ew = (const float*)d_in[2];
    const float* Wm = (const float*)d_in[3];    // [4, 128, 128]
    const float* bm = (const float*)d_in[4];    // [4, 128]
    const float* gm = (const float*)d_in[5];
    const float* bt = (const float*)d_in[6];
    float* out = (float*)d_out;

    const int N = in_sizes[0] / DF;
    const int E = in_sizes[2];
    const int ND = N * DF;

    // Carve workspace
    float* ws    = (float*)d_ws;
    float* h     = ws;  ws += ND;
    float* agg   = ws;  ws += ND;
    float* xbuf  = ws;  ws += ND;
    float* deg   = ws;  ws += N;
    float* dis   = ws;  ws += N;
    float* selfn = ws;  ws += N;
    float* norm  = ws;  ws += E;
    float* gsum  = ws;  ws += 128;
    float* gsq   = ws;  ws += 128;
    float* scale = ws;  ws += 128;
    float* shift = ws;  ws += 128;

    const int* srcI = ei;
    const int* dstI = ei + E;

    const int TB = 256;
    auto cdiv = [](int a, int b) { return (a + b - 1) / b; };

    // gcn_norm
    k_deg_init <<<cdiv(N, TB), TB, 0, stream>>>(deg, N);
    k_deg_edges<<<cdiv(E, TB), TB, 0, stream>>>(dstI, ew, deg, E);
    k_dis      <<<cdiv(N, TB), TB, 0, stream>>>(deg, dis, selfn, N);
    k_norm     <<<cdiv(E, TB), TB, 0, stream>>>(srcI, dstI, ew, dis, norm, E);

    const int rowTiles = cdiv(N, 128);
    const int edgeThreads = E * 32;

    for (int i = 0; i < 4; ++i) {
        const float* xin = (i == 0) ? x0 : xbuf;
        float* xout = (i == 3) ? out : xbuf;

        k_gemm_wmma<<<rowTiles, 256, 0, stream>>>(xin, Wm + (size_t)i * DF * DF, h, N);
        k_agg_init <<<cdiv(ND, TB), TB, 0, stream>>>(h, selfn, agg, ND);
        k_agg_edges<<<cdiv(edgeThreads, TB), TB, 0, stream>>>(srcI, dstI, norm, h, agg, E);

        if (i != 2) {   // BN layers: 0, 1, 3
            k_zero_stats <<<1, 128, 0, stream>>>(gsum, gsq);
            k_post_z     <<<1024, 256, 0, stream>>>(agg, bm + i * DF, x0, h,
                                                    gsum, gsq, ND, (i < 3) ? 1 : 0);
            k_bn_finalize<<<1, 128, 0, stream>>>(gsum, gsq, gm + i * DF, bt + i * DF,
                                                 scale, shift, N);
            k_bn_apply   <<<cdiv(ND, TB), TB, 0, stream>>>(h, scale, shift, xout, ND);
        } else {
            k_post_nobn  <<<cdiv(ND, TB), TB, 0, stream>>>(agg, bm + i * DF, x0, xout, ND);
        }
    }
}